// LierePositionEncoder_11854109737108
// MI455X (gfx1250) — compile-verified
//
#include <hip/hip_runtime.h>

// LIERE position encoder: out[n] = expm(x_n*S0 + y_n*S1), S skew-symmetric 128x128.
// Scaling (2^-12) + order-8 Taylor (Horner) + 12 squarings, all matmuls via
// V_WMMA_F32_16X16X4_F32 (fp32 WMMA -- fp16/bf16 would lose ~2^12 x rounding
// error through the squaring chain).

typedef __attribute__((ext_vector_type(2))) float v2f;
typedef __attribute__((ext_vector_type(8))) float v8f;

#define D       128
#define LDP     132                 // padded LDS row stride (floats): bank advance 4/row
#define STEPS   37
#define NPOS    (STEPS * STEPS)     // 1369
#define NOUT    (NPOS + 1)          // + CLS
#define KTERMS  8                   // Taylor order
#define SSHIFT  12                  // scaling: G / 2^12

__device__ __forceinline__ v8f wmma4(v2f a, v2f b, v8f c) {
  // D = A(16x4, f32) * B(4x16, f32) + C(16x16, f32)
  return __builtin_amdgcn_wmma_f32_16x16x4_f32(
      /*neg_a=*/false, a, /*neg_b=*/false, b,
      /*c_mod=*/(short)0, c, /*reuse_a=*/false, /*reuse_b=*/false);
}

__global__ __launch_bounds__(256)
void liere_expm_kernel(const float* __restrict__ gen, float* __restrict__ out) {
  extern __shared__ float smem[];
  float* bufA = smem;                 // R ping
  float* bufB = smem + D * LDP;       // R pong
  float* Gm   = smem + 2 * D * LDP;   // scaled generator

  const int blk = blockIdx.x;
  const int t   = threadIdx.x;
  float* dst = out + (size_t)blk * (D * D);

  if (blk == 0) {                     // CLS token: expm(0) = I
    for (int e = t; e < D * D; e += 256) {
      int i = e >> 7, j = e & 127;
      dst[e] = (i == j) ? 1.0f : 0.0f;
    }
    return;
  }

  // Warm the 128KB of generator params into cache (global_prefetch_b8).
  if (t < 64) __builtin_prefetch(gen + (size_t)t * 512, 0, 0);

  const int   p    = blk - 1;
  const float x    = (float)(p / STEPS) * (1.0f / (float)(STEPS - 1));
  const float y    = (float)(p % STEPS) * (1.0f / (float)(STEPS - 1));
  const float sc   = 1.0f / (float)(1 << SSHIFT);
  const float invK = 1.0f / (float)KTERMS;

  const float* g0 = gen;
  const float* g1 = gen + D * D;

  // Build G = sc*(x*S0 + y*S1) (skew on the fly; triu(g)-0.5 cancels in skew)
  // and R0 = I + G/K (folds the innermost Horner step).
  for (int e = t; e < D * D; e += 256) {
    int i = e >> 7, j = e & 127;
    float s0 = 0.0f, s1 = 0.0f;
    if (i < j)      { s0 =  g0[i * D + j]; s1 =  g1[i * D + j]; }
    else if (i > j) { s0 = -g0[j * D + i]; s1 = -g1[j * D + i]; }
    float gv = sc * (x * s0 + y * s1);
    Gm[i * LDP + j]   = gv;
    bufA[i * LDP + j] = gv * invK + ((i == j) ? 1.0f : 0.0f);
  }
  __syncthreads();

  const int w  = t >> 5;        // wave id (wave32)
  const int ln = t & 31;
  const int lm = ln & 15;
  const int hi = ln >> 4;       // half-wave select (K split per ISA 7.12.2)
  const int mb = (w & 3) * 32;  // this wave: rows [mb, mb+32)
  const int nb = (w >> 2) * 64; //            cols [nb, nb+64)

  // 19 phases: 7 Horner steps (R <- I + (G*R)/k, k=7..1), then 12 squarings.
  for (int ph = 0; ph < 19; ++ph) {
    const bool   horner = (ph < KTERMS - 1);
    const float* Ym = (ph & 1) ? bufB : bufA;          // right operand / prev R
    const float* Xm = horner ? Gm : Ym;                // left operand
    float*       Zm = (ph & 1) ? bufA : bufB;          // result
    const float  scl = horner ? 1.0f / (float)(KTERMS - 1 - ph) : 1.0f;

    v8f acc[2][4];
#pragma unroll
    for (int mi = 0; mi < 2; ++mi)
#pragma unroll
      for (int ni = 0; ni < 4; ++ni)
        acc[mi][ni] = (v8f){0.f, 0.f, 0.f, 0.f, 0.f, 0.f, 0.f, 0.f};

    for (int k = 0; k < D; k += 4) {
      // A frags: lane (lm,hi) holds X[m0+lm][k+2*hi .. k+2*hi+1]
      v2f af0 = *(const v2f*)&Xm[(mb +      lm) * LDP + k + 2 * hi];
      v2f af1 = *(const v2f*)&Xm[(mb + 16 + lm) * LDP + k + 2 * hi];
      // B frags: lane (lm,hi) holds Y[k+2*hi][n0+lm], Y[k+2*hi+1][n0+lm]
      v2f bf[4];
      const int kk = k + 2 * hi;
#pragma unroll
      for (int ni = 0; ni < 4; ++ni) {
        const int nc = nb + ni * 16 + lm;
        bf[ni].x = Ym[kk * LDP + nc];
        bf[ni].y = Ym[(kk + 1) * LDP + nc];
      }
#pragma unroll
      for (int ni = 0; ni < 4; ++ni) {
        acc[0][ni] = wmma4(af0, bf[ni], acc[0][ni]);
        acc[1][ni] = wmma4(af1, bf[ni], acc[1][ni]);
      }
    }

    // Store D tiles: element r of lane (lm,hi) -> row m0+r+8*hi, col n0+lm.
#pragma unroll
    for (int mi = 0; mi < 2; ++mi) {
#pragma unroll
      for (int ni = 0; ni < 4; ++ni) {
        const int nc = nb + ni * 16 + lm;
#pragma unroll
        for (int r = 0; r < 8; ++r) {
          const int row = mb + mi * 16 + r + 8 * hi;
          float v = acc[mi][ni][r] * scl;
          if (horner && row == nc) v += 1.0f;
          Zm[row * LDP + nc] = v;
        }
      }
    }
    __syncthreads();
  }

  // Phase 18 (even) wrote bufB. Coalesced copy LDS -> HBM.
  for (int e = t; e < D * D; e += 256) {
    int i = e >> 7, j = e & 127;
    dst[e] = bufB[i * LDP + j];
  }
}

extern "C" void kernel_launch(void* const* d_in, const int* in_sizes, int n_in,
                              void* d_out, int out_size, void* d_ws, size_t ws_size,
                              hipStream_t stream) {
  (void)in_sizes; (void)n_in; (void)d_ws; (void)ws_size; (void)out_size;
  // d_in[0] = image_sizes (int32, unused: sizes are static per the reference)
  // d_in[1] = generator_raw_params [2,1,128,128] f32
  const float* gen = (const float*)d_in[1];
  float* out = (float*)d_out;
  const size_t shmem = 3u * D * LDP * sizeof(float);   // 198 KB < 320 KB/WGP
  liere_expm_kernel<<<NOUT, 256, shmem, stream>>>(gen, out);
}